// MLStruct_90881507983367
// MI455X (gfx1250) — compile-verified
//
#include <hip/hip_runtime.h>
#include <stdint.h>

#define N_LAYERS 2
#define N_NODES  50000
#define N_EDGES  800000
#define DIM      256
#define MPAD     50016   // N_NODES rounded up to multiple of 32

typedef float v2f __attribute__((ext_vector_type(2)));
typedef float v8f __attribute__((ext_vector_type(8)));
typedef int   ai4 __attribute__((ext_vector_type(4)));
typedef __attribute__((address_space(1))) ai4* g_ai4;   // global src for async-to-LDS
typedef __attribute__((address_space(3))) ai4* l_ai4;   // LDS dst for async-to-LDS

#if defined(__has_builtin)
#  if __has_builtin(__builtin_amdgcn_global_load_async_to_lds_b128)
#    define HAVE_ASYNC_LDS 1
#  endif
#endif
#ifndef HAVE_ASYNC_LDS
#  define HAVE_ASYNC_LDS 0
#endif

// ---------------------------------------------------------------------------
// zero a float buffer (float4-wide)
// ---------------------------------------------------------------------------
__global__ __launch_bounds__(256) void zero_f4(float4* __restrict__ p, int n4) {
    int i = blockIdx.x * 256 + threadIdx.x;
    if (i < n4) p[i] = make_float4(0.f, 0.f, 0.f, 0.f);
}

// ---------------------------------------------------------------------------
// Edge scatter: S[dst[e], j] += ReLU(ew[e]*W1[j] + b1[j]);  degf[dst[e]] += 1
// 256 threads = 256 features; 128 edges staged in LDS per block.
// Wave lanes j..j+31 hit 32 consecutive floats of one node row -> 128B atomic
// segments. Atomic skipped when ReLU output is zero (~50% of adds).
// ---------------------------------------------------------------------------
__global__ __launch_bounds__(256)
void edge_scatter(const long long* __restrict__ dst,
                  const float* __restrict__ ew,
                  const float* __restrict__ W1,
                  const float* __restrict__ b1,
                  float* __restrict__ S,
                  float* __restrict__ degf) {
    __shared__ int   s_dst[128];
    __shared__ float s_ew[128];
    const int tid = threadIdx.x;
    const long long e0 = (long long)blockIdx.x * 128;
    if (tid < 128) {
        long long e = e0 + tid;
        int d = (int)dst[e];
        s_dst[tid] = d;
        s_ew[tid]  = ew[e];
        unsafeAtomicAdd(&degf[d], 1.0f);
    }
    __syncthreads();
    const float w = W1[tid];
    const float b = b1[tid];
#pragma unroll 4
    for (int i = 0; i < 128; ++i) {
        float g = fmaf(s_ew[i], w, b);
        if (g > 0.0f)
            unsafeAtomicAdd(&S[(size_t)s_dst[i] * DIM + tid], g);
    }
}

// ---------------------------------------------------------------------------
// C = A[rows,256] @ B[256,256] + epilogue, using V_WMMA_F32_16X16X4_F32.
//   MODE 0: C = A@B + degf[row]*bias[col]          (no store guard)
//   MODE 1: C = ReLU(A@B + bias[col])              (no store guard)
//   MODE 2: C = A@B + bias[col], rows < m_store    (guard, last block only)
// Block = 256 threads (8 waves), tile = 32 rows x 256 cols.
// Wave w: rows 16*(w/4).., cols 64*(w%4)..  (4 accumulators of 16x16).
// A staged in LDS (async-to-LDS when available), row stride 260 floats
// (16B-aligned float4 slots, conflict-free 8B WMMA-A reads).
// Safe in-place (each block reads/writes only its own 32 rows).
// ---------------------------------------------------------------------------
template <int MODE>
__global__ __launch_bounds__(256)
void gemm256(const float* __restrict__ A,
             const float* __restrict__ B,
             const float* __restrict__ bias,
             const float* __restrict__ degf,
             float* __restrict__ C,
             int m_store) {
    __shared__ float As[32 * 260];
    const int tid  = threadIdx.x;
    const int row0 = blockIdx.x * 32;

    // warm B (256KB, shared by all blocks) into near caches: 8 lines/thread
#pragma unroll
    for (int i = 0; i < 8; ++i)
        __builtin_prefetch(&B[(tid * 8 + i) * 32], 0, 1);

    // stage 32x256 A tile into LDS
#if HAVE_ASYNC_LDS
    for (int idx = tid; idx < 32 * 64; idx += 256) {
        int r = idx >> 6, c4 = (idx & 63) << 2;
        float* gp = const_cast<float*>(A) + (size_t)(row0 + r) * DIM + c4;
        __builtin_amdgcn_global_load_async_to_lds_b128(
            (g_ai4)gp, (l_ai4)&As[r * 260 + c4], 0, 0);
    }
#  if __has_builtin(__builtin_amdgcn_s_wait_asynccnt)
    __builtin_amdgcn_s_wait_asynccnt(0);
#  else
    asm volatile("s_wait_asynccnt 0x0" ::: "memory");
#  endif
#else
    for (int idx = tid; idx < 32 * 64; idx += 256) {
        int r = idx >> 6, c4 = (idx & 63) << 2;
        float4 v = *(const float4*)(A + (size_t)(row0 + r) * DIM + c4);
        *(float4*)(&As[r * 260 + c4]) = v;
    }
#endif
    __syncthreads();

    const int wave = tid >> 5, lane = tid & 31;
    const int half = lane >> 4, l16 = lane & 15;
    const int wr0 = 16 * (wave >> 2);   // 0 or 16
    const int wn0 = 64 * (wave & 3);    // 0,64,128,192

    v8f acc[4] = {};
    const float* Arow = &As[(wr0 + l16) * 260];

    for (int k0 = 0; k0 < DIM; k0 += 4) {
        // A 16x4 frag: vgpr0={K=k0 | K=k0+2}, vgpr1={K=k0+1 | K=k0+3}, M=lane%16
        v2f a = *(const v2f*)(Arow + k0 + 2 * half);
#pragma unroll
        for (int ct = 0; ct < 4; ++ct) {
            const int col = wn0 + 16 * ct + l16;
            v2f bf;
            bf.x = B[(size_t)(k0 + 2 * half) * DIM + col];
            bf.y = B[(size_t)(k0 + 2 * half + 1) * DIM + col];
            acc[ct] = __builtin_amdgcn_wmma_f32_16x16x4_f32(
                false, a, false, bf, (short)0, acc[ct], false, false);
        }
    }

    // epilogue: C/D layout -> lane: N=lane%16, vgpr v: M = v + 8*(lane/16)
    float bias_c[4];
#pragma unroll
    for (int ct = 0; ct < 4; ++ct)
        bias_c[ct] = bias[wn0 + 16 * ct + l16];

#pragma unroll
    for (int v = 0; v < 8; ++v) {
        const int row = row0 + wr0 + 8 * half + v;
        if (MODE == 2 && row >= m_store) continue;   // dead code for MODE 0/1
        const float dg = (MODE == 0) ? degf[row] : 0.0f;
        float* crow = C + (size_t)row * DIM;
#pragma unroll
        for (int ct = 0; ct < 4; ++ct) {
            float val = acc[ct][v];
            if (MODE == 0)      val = fmaf(dg, bias_c[ct], val);
            else if (MODE == 1) val = fmaxf(val + bias_c[ct], 0.0f);
            else                val += bias_c[ct];
            crow[wn0 + 16 * ct + l16] = val;
        }
    }
}

// ---------------------------------------------------------------------------
extern "C" void kernel_launch(void* const* d_in, const int* in_sizes, int n_in,
                              void* d_out, int out_size, void* d_ws, size_t ws_size,
                              hipStream_t stream) {
    const long long* edge_index = (const long long*)d_in[0]; // [2,2,E] int64
    const float* edge_weight = (const float*)d_in[1];        // [2,E]
    const float* We1 = (const float*)d_in[2];                // [2,1,256]
    const float* be1 = (const float*)d_in[3];                // [2,256]
    const float* We2 = (const float*)d_in[4];                // [2,256,256]
    const float* be2 = (const float*)d_in[5];                // [2,256]
    const float* Wn1 = (const float*)d_in[6];                // [2,256,256]
    const float* bn1 = (const float*)d_in[7];                // [2,256]
    const float* Wn2 = (const float*)d_in[8];                // [2,256,256]
    const float* bn2 = (const float*)d_in[9];                // [2,256]
    float* out = (float*)d_out;                              // [2,50000,256]

    float* S    = (float*)d_ws;                // MPAD*256 floats (~51.2 MB)
    float* degf = S + (size_t)MPAD * DIM;      // MPAD floats

    for (int l = 0; l < N_LAYERS; ++l) {
        const long long* dst = edge_index + (size_t)l * 2 * N_EDGES + N_EDGES;
        const float* ew = edge_weight + (size_t)l * N_EDGES;

        zero_f4<<<(MPAD * DIM / 4) / 256, 256, 0, stream>>>((float4*)S, MPAD * DIM / 4);
        zero_f4<<<(MPAD / 4 + 255) / 256, 256, 0, stream>>>((float4*)degf, MPAD / 4);

        edge_scatter<<<N_EDGES / 128, 256, 0, stream>>>(
            dst, ew, We1 + (size_t)l * DIM, be1 + (size_t)l * DIM, S, degf);

        // Y1 = S@We2 + deg*be2   (in-place on S)
        gemm256<0><<<MPAD / 32, 256, 0, stream>>>(
            S, We2 + (size_t)l * DIM * DIM, be2 + (size_t)l * DIM, degf, S, MPAD);
        // Y2 = ReLU(Y1@Wn1 + bn1)   (in-place on S)
        gemm256<1><<<MPAD / 32, 256, 0, stream>>>(
            S, Wn1 + (size_t)l * DIM * DIM, bn1 + (size_t)l * DIM, nullptr, S, MPAD);
        // out = Y2@Wn2 + bn2
        gemm256<2><<<MPAD / 32, 256, 0, stream>>>(
            S, Wn2 + (size_t)l * DIM * DIM, bn2 + (size_t)l * DIM, nullptr,
            out + (size_t)l * N_NODES * DIM, N_NODES);
    }
}